// ResidualVectorQuantizer_66468913873595
// MI455X (gfx1250) — compile-verified
//
#include <hip/hip_runtime.h>
#include <hip/hip_bf16.h>
#include <math.h>

#define LATENT 128
#define NQ 8
#define KCB 1024
#define BATCH 8
#define SEQ 4096
#define NVEC (BATCH * SEQ)      // 32768
#define MT 128                  // rows per workgroup
#define NWG (NVEC / MT)         // 256
#define CHUNK 16                // codebook columns per tile
#define NCHUNK (KCB / CHUNK)    // 64
#define DSTEPS (LATENT / 4)     // 32 wmma k-steps
#define LDS_STRIDE 132          // padded row stride (floats) to avoid bank conflicts
#define TILE_F (CHUNK * LDS_STRIDE)

typedef __attribute__((ext_vector_type(2))) float v2f;
typedef __attribute__((ext_vector_type(8))) float v8f;

// ---------------------------------------------------------------------------
// Kernel 1: half squared norms of every code: hn[q*K + k] = 0.5 * ||c_qk||^2
// ---------------------------------------------------------------------------
__global__ void rvq_halfnorm(const float* __restrict__ cb, float* __restrict__ hn) {
    int k = blockIdx.x * blockDim.x + threadIdx.x;   // 0 .. NQ*KCB-1
    const float* c = cb + (size_t)k * LATENT;
    float s = 0.f;
#pragma unroll 8
    for (int d = 0; d < LATENT; ++d) s += c[d] * c[d];
    hn[k] = 0.5f * s;
}

// ---------------------------------------------------------------------------
// Async copy of one 16x128 codebook tile (8 KB) into LDS, 32 B per thread.
// The offset: immediate is added to BOTH lds and global addresses (ISA §08).
// ---------------------------------------------------------------------------
__device__ __forceinline__ void async_tile_load(float* ldsbase,
                                                const float* gsrc, int tid) {
    int c16 = tid >> 4;          // code row 0..15
    int dq  = (tid & 15) * 8;    // dim start (8 floats = 32 B per thread)
    const float* g = gsrc + (size_t)c16 * LATENT + dq;
    unsigned lds_off = (unsigned)(size_t)(ldsbase + c16 * LDS_STRIDE + dq);
    unsigned long long ga = (unsigned long long)(size_t)g;
    asm volatile(
        "global_load_async_to_lds_b128 %0, %1, off\n\t"
        "global_load_async_to_lds_b128 %0, %1, off offset:16"
        :: "v"(lds_off), "v"(ga) : "memory");
}

// ---------------------------------------------------------------------------
// Kernel 2: fused 8-stage RVQ. Residual kept in registers as WMMA A-fragments.
// ---------------------------------------------------------------------------
__global__ __launch_bounds__(256) void rvq_main(
    const float* __restrict__ z, const float* __restrict__ cb,
    const float* __restrict__ hn, float* __restrict__ out_q,
    float* __restrict__ out_idx, unsigned* __restrict__ hist,
    float* __restrict__ losspart)
{
    __shared__ float tile[3][TILE_F];   // triple-buffered 16x128 tiles
    __shared__ int   selbuf[8][16];     // per-wave selected code per row
    __shared__ float red[8];            // per-wave loss partials

    const int tid  = threadIdx.x;
    const int wave = tid >> 5;
    const int lane = tid & 31;
    const int col  = lane & 15;        // B-matrix column / C-matrix N index
    const int hi   = lane >> 4;        // 0: lanes 0-15, 1: lanes 16-31
    const int off  = hi * 2;           // K%4 offset this lane holds {off, off+1}

    const int nbase = blockIdx.x * MT + wave * 16;
    const int m = col;                 // A-matrix row this lane carries
    const int n = nbase + m;           // global vector id
    const int b = n >> 12;             // n / SEQ
    const int s = n & (SEQ - 1);       // n % SEQ

    // ---- load residual = z (gather from [B, D, S]) into A-fragments -------
    v2f a[DSTEPS];
    {
        const float* zp = z + ((size_t)b * LATENT) * SEQ + s;
#pragma unroll
        for (int t = 0; t < DSTEPS; ++t) {
            int d = 4 * t + off;
            a[t].x = zp[(size_t)d * SEQ];
            a[t].y = zp[(size_t)(d + 1) * SEQ];
        }
    }

    for (int q = 0; q < NQ; ++q) {
        const float* cbq = cb + (size_t)q * KCB * LATENT;
        const float* hnq = hn + q * KCB;

        float bestv[8];
        int   besti[8];
#pragma unroll
        for (int j = 0; j < 8; ++j) { bestv[j] = -3.4e38f; besti[j] = 0; }

        // software pipeline: prefetch chunk 0, then 1 barrier per chunk
        async_tile_load(tile[0], cbq, tid);

        for (int ch = 0; ch < NCHUNK; ++ch) {
            const int kbase = ch * CHUNK;
            const float* cur = tile[ch % 3];

            if (ch + 1 < NCHUNK) {
                async_tile_load(tile[(ch + 1) % 3],
                                cbq + (size_t)(ch + 1) * CHUNK * LATENT, tid);
                asm volatile("s_wait_asynccnt 2" ::: "memory");
            } else {
                asm volatile("s_wait_asynccnt 0" ::: "memory");
            }
            __syncthreads();   // all waves' async writes for `cur` visible

            float hnl = hnq[kbase + col];

            // ---- 16x16 dot-product tile via f32 WMMA, K = 128 -------------
            v8f acc = {};
#pragma unroll
            for (int t = 0; t < DSTEPS; ++t) {
                v2f bf = *(const v2f*)&cur[col * LDS_STRIDE + 4 * t + off];
                acc = __builtin_amdgcn_wmma_f32_16x16x4_f32(
                    false, a[t], false, bf, (short)0, acc, false, false);
            }

            // ---- argmax(dot - 0.5||c||^2) == argmin distance --------------
#pragma unroll
            for (int j = 0; j < 8; ++j) {
                float v  = acc[j] - hnl;
                int   ix = kbase + col;
#pragma unroll
                for (int sh = 1; sh < 16; sh <<= 1) {
                    float ov = __shfl_xor(v, sh, 16);
                    int   oi = __shfl_xor(ix, sh, 16);
                    if (ov > v || (ov == v && oi < ix)) { v = ov; ix = oi; }
                }
                if (v > bestv[j] || (v == bestv[j] && ix < besti[j])) {
                    bestv[j] = v; besti[j] = ix;
                }
            }
        }

        // ---- publish per-row selections (lane0: rows 0-7, lane16: 8-15) ---
        if (lane == 0) {
#pragma unroll
            for (int j = 0; j < 8; ++j) selbuf[wave][j] = besti[j];
        } else if (lane == 16) {
#pragma unroll
            for (int j = 0; j < 8; ++j) selbuf[wave][8 + j] = besti[j];
        }
        __syncthreads();

        // ---- emit indices + histogram -------------------------------------
        if (lane == 0 || lane == 16) {
            int rbase = hi * 8;
#pragma unroll
            for (int j = 0; j < 8; ++j) {
                int row = rbase + j;
                int idx = selbuf[wave][row];
                out_idx[(size_t)(nbase + row) * NQ + q] = (float)idx;
                atomicAdd(&hist[q * KCB + idx], 1u);
            }
        }

        // ---- residual update: r -= selected code --------------------------
        {
            int sel = selbuf[wave][m];
            const float* cp = cbq + (size_t)sel * LATENT + off;
#pragma unroll
            for (int t = 0; t < DSTEPS; ++t) {
                v2f cv = *(const v2f*)(cp + 4 * t);
                a[t].x -= cv.x;
                a[t].y -= cv.y;
            }
        }
        __syncthreads();   // selbuf + tile buffers safe for next stage
    }

    // ---- commitment loss partial: sum residual^2 ---------------------------
    float ls = 0.f;
#pragma unroll
    for (int t = 0; t < DSTEPS; ++t) ls += a[t].x * a[t].x + a[t].y * a[t].y;
#pragma unroll
    for (int sh = 1; sh < 32; sh <<= 1) ls += __shfl_xor(ls, sh, 32);
    if (lane == 0) red[wave] = ls;
    __syncthreads();
    if (tid == 0) {
        float t = 0.f;
#pragma unroll
        for (int w = 0; w < 8; ++w) t += red[w];
        losspart[blockIdx.x] = t;
    }

    // ---- quantized_st == quantized = z - residual_final, scatter to [B,D,S]
    {
        const float* zp = z + ((size_t)b * LATENT) * SEQ + s;
        float* op = out_q + ((size_t)b * LATENT) * SEQ + s;
#pragma unroll
        for (int t = 0; t < DSTEPS; ++t) {
            int d = 4 * t + off;
            op[(size_t)d * SEQ]       = zp[(size_t)d * SEQ]       - a[t].x;
            op[(size_t)(d + 1) * SEQ] = zp[(size_t)(d + 1) * SEQ] - a[t].y;
        }
    }
}

// ---------------------------------------------------------------------------
// Kernel 3: deterministic finalize (loss mean + perplexity from histogram)
// ---------------------------------------------------------------------------
__global__ void rvq_finalize(const float* __restrict__ losspart,
                             const unsigned* __restrict__ hist,
                             float* __restrict__ out_scalars)
{
    __shared__ float red[256];
    int tid = threadIdx.x;

    red[tid] = losspart[tid];                      // 256 partials
    __syncthreads();
    for (int s = 128; s > 0; s >>= 1) {
        if (tid < s) red[tid] += red[tid + s];
        __syncthreads();
    }
    float loss = red[0] / (float)((size_t)NVEC * LATENT);
    __syncthreads();

    float ppl = 0.f;
    for (int q = 0; q < NQ; ++q) {
        float e = 0.f;
        for (int k = tid; k < KCB; k += 256) {
            float p = (float)hist[q * KCB + k] / (float)NVEC;
            e -= p * logf(p + 1e-10f);
        }
        red[tid] = e;
        __syncthreads();
        for (int s = 128; s > 0; s >>= 1) {
            if (tid < s) red[tid] += red[tid + s];
            __syncthreads();
        }
        if (tid == 0) ppl += expf(red[0]);
        __syncthreads();
    }
    if (tid == 0) {
        out_scalars[0] = loss;
        out_scalars[1] = ppl / (float)NQ;
    }
}

// ---------------------------------------------------------------------------
// d_ws layout (4-byte units): [0,8192) halfnorm | [8192,16384) hist (uint)
//                             [16384,16640) per-WG loss partials
// d_out: quantized [8*128*4096] | indices-as-f32 [32768*8] | loss | perplexity
// ---------------------------------------------------------------------------
extern "C" void kernel_launch(void* const* d_in, const int* in_sizes, int n_in,
                              void* d_out, int out_size, void* d_ws, size_t ws_size,
                              hipStream_t stream) {
    const float* z  = (const float*)d_in[0];
    const float* cb = (const float*)d_in[1];

    float*    hn       = (float*)d_ws;
    unsigned* hist     = (unsigned*)d_ws + NQ * KCB;
    float*    losspart = (float*)d_ws + 2 * NQ * KCB;

    float* out_q       = (float*)d_out;
    float* out_idx     = out_q + (size_t)BATCH * LATENT * SEQ;
    float* out_scalars = out_idx + (size_t)NVEC * NQ;

    // zero histogram + loss partials every launch (graph-capture safe)
    hipMemsetAsync((char*)d_ws + (size_t)NQ * KCB * 4, 0,
                   (size_t)(NQ * KCB + NWG) * 4, stream);

    rvq_halfnorm<<<(NQ * KCB) / 256, 256, 0, stream>>>(cb, hn);
    rvq_main<<<NWG, 256, 0, stream>>>(z, cb, hn, out_q, out_idx, hist, losspart);
    rvq_finalize<<<1, 256, 0, stream>>>(losspart, hist, out_scalars);
}